// ResidualSeparableBlock_71820443124033
// MI455X (gfx1250) — compile-verified
//
#include <hip/hip_runtime.h>

#define N_PIX 196608
#define FCH 32

typedef float v2f __attribute__((ext_vector_type(2)));
typedef float v8f __attribute__((ext_vector_type(8)));

// ---------------------------------------------------------------------------
// Kernel 1: T1 = L(T0) (fixed-degree gather) ; Y = wd[0] (x) T0 + wd[1] (x) T1
// One wave per node, lane = feature channel. idx/wnb loads are wave-uniform
// (scalar load path), neighbor-row loads are coalesced 128B.
// ---------------------------------------------------------------------------
template <int NB>
__global__ void cheb_first_kernel(const float* __restrict__ T0,
                                  const int* __restrict__ idx,
                                  const float* __restrict__ wnb,
                                  const float* __restrict__ wd,
                                  float* __restrict__ T1,
                                  float* __restrict__ Y) {
    int n = (int)((blockIdx.x * blockDim.x + threadIdx.x) >> 5);
    int f = threadIdx.x & 31;
    if (n >= N_PIX) return;
    const int*   ip = idx + (long)n * NB;
    const float* wp = wnb + (long)n * NB;
    float acc = 0.f;
#pragma unroll
    for (int j = 0; j < NB; ++j) {
        int m = ip[j];
        acc = fmaf(wp[j], T0[(long)m * FCH + f], acc);
    }
    T1[(long)n * FCH + f] = acc;
    float t0 = T0[(long)n * FCH + f];
    v2f yv;
    yv.x = wd[(0 * FCH + f) * 2 + 0] * t0 + wd[(1 * FCH + f) * 2 + 0] * acc;
    yv.y = wd[(0 * FCH + f) * 2 + 1] * t0 + wd[(1 * FCH + f) * 2 + 1] * acc;
    *(v2f*)(Y + (long)n * (2 * FCH) + 2 * f) = yv;
}

// ---------------------------------------------------------------------------
// Kernel 2: Tnew = 2*L(G) - P ; D = Tnew (D may alias P: same-element RAW only)
//           Y[n, 2f+d] += wd[k,f,d] * Tnew
// ---------------------------------------------------------------------------
template <int NB>
__global__ void cheb_step_kernel(const float* __restrict__ G,
                                 const float* P,
                                 float* D,
                                 const int* __restrict__ idx,
                                 const float* __restrict__ wnb,
                                 const float* __restrict__ wd, int k,
                                 float* __restrict__ Y) {
    int n = (int)((blockIdx.x * blockDim.x + threadIdx.x) >> 5);
    int f = threadIdx.x & 31;
    if (n >= N_PIX) return;
    const int*   ip = idx + (long)n * NB;
    const float* wp = wnb + (long)n * NB;
    float acc = 0.f;
#pragma unroll
    for (int j = 0; j < NB; ++j) {
        int m = ip[j];
        acc = fmaf(wp[j], G[(long)m * FCH + f], acc);
    }
    float tnew = 2.f * acc - P[(long)n * FCH + f];
    D[(long)n * FCH + f] = tnew;
    v2f yv = *(v2f*)(Y + (long)n * (2 * FCH) + 2 * f);
    yv.x = fmaf(wd[(k * FCH + f) * 2 + 0], tnew, yv.x);
    yv.y = fmaf(wd[(k * FCH + f) * 2 + 1], tnew, yv.y);
    *(v2f*)(Y + (long)n * (2 * FCH) + 2 * f) = yv;
}

// ---------------------------------------------------------------------------
// Kernel 3: pointwise GEMM [16 nodes x 64] @ [64 x 32] via v_wmma_f32_16x16x4,
// fused bias + relu + layernorm (+ residual). One wave handles 16 nodes; two
// N-tiles (cols 0-15, 16-31); 16 K-steps of depth 4.
//
// Wp is staged in LDS pre-interleaved as K-pairs:
//   wlds[(k>>1)*64 + c*2 + (k&1)] = Wp[k*32 + c]
// so a B fragment {Wp[k0][c], Wp[k0+1][c]} is one contiguous 8B LDS load, and
// the two N-tile fragments (c = mrow, c = 16+mrow) are 128B apart, which the
// compiler can fuse into a single ds_load_2addr_b64 feeding WMMA directly.
//
// A 16x4 f32 layout: lanes 0-15 rows M=0..15 holding K={k0,k0+1},
//                    lanes 16-31 same rows holding K={k0+2,k0+3}.
// C/D: VGPR r, lanes 0-15 => M=r, N=lane; lanes 16-31 => M=r+8, N=lane-16.
// Grid exactly covers N_PIX => EXEC all ones around WMMA (required).
// ---------------------------------------------------------------------------
__global__ void pointwise_ln_kernel(const float* __restrict__ Y,
                                    const float* __restrict__ Wp,
                                    const float* __restrict__ bias,
                                    const float* __restrict__ gamma,
                                    const float* __restrict__ beta,
                                    const float* __restrict__ residual,
                                    float* __restrict__ out) {
    __shared__ float wlds[64 * 32];   // interleaved: [32 k-pairs][32 cols][2]
    for (int i = threadIdx.x; i < 64 * 32; i += blockDim.x) {
        int k = i >> 5;
        int c = i & 31;
        wlds[(k >> 1) * 64 + c * 2 + (k & 1)] = Wp[i];
    }
    __syncthreads();

    int waveInBlock = threadIdx.x >> 5;
    int lane = threadIdx.x & 31;
    int hi = lane >> 4;       // lane half
    int mrow = lane & 15;
    long base = (long)(blockIdx.x * 8 + waveInBlock) * 16;  // first node of tile

    v8f c0 = {};
    v8f c1 = {};
    const float* arow = Y + (base + mrow) * 64;
    const float* brow = wlds + hi * 64 + mrow * 2;
#pragma unroll
    for (int kk = 0; kk < 16; ++kk) {
        int k0 = kk * 4 + 2 * hi;
        v2f a  = *(const v2f*)(arow + k0);
        v2f b0 = *(const v2f*)(brow + kk * 128);
        v2f b1 = *(const v2f*)(brow + kk * 128 + 32);
        c0 = __builtin_amdgcn_wmma_f32_16x16x4_f32(false, a, false, b0,
                                                   (short)0, c0, false, false);
        c1 = __builtin_amdgcn_wmma_f32_16x16x4_f32(false, a, false, b1,
                                                   (short)0, c1, false, false);
    }

    // bias + relu
    float v0[8], v1[8];
    float bia0 = bias[mrow], bia1 = bias[16 + mrow];
#pragma unroll
    for (int r = 0; r < 8; ++r) {
        v0[r] = fmaxf(c0[r] + bia0, 0.f);
        v1[r] = fmaxf(c1[r] + bia1, 0.f);
    }

    // layernorm per node: node m's 32 features live across one 16-lane half
    float ga0 = gamma[mrow], ga1 = gamma[16 + mrow];
    float be0 = beta[mrow],  be1 = beta[16 + mrow];
#pragma unroll
    for (int r = 0; r < 8; ++r) {
        float s = v0[r] + v1[r];
        float q = v0[r] * v0[r] + v1[r] * v1[r];
#pragma unroll
        for (int m = 1; m < 16; m <<= 1) {  // stays within 16-lane half
            s += __shfl_xor(s, m, 32);
            q += __shfl_xor(q, m, 32);
        }
        float mu = s * (1.f / 32.f);
        float var = q * (1.f / 32.f) - mu * mu;
        float rstd = rsqrtf(var + 1e-6f);
        long node = base + r + 8 * hi;
        float o0 = (v0[r] - mu) * rstd * ga0 + be0;
        float o1 = (v1[r] - mu) * rstd * ga1 + be1;
        if (residual) {
            o0 += residual[node * 32 + mrow];
            o1 += residual[node * 32 + 16 + mrow];
        }
        out[node * 32 + mrow] = o0;
        out[node * 32 + 16 + mrow] = o1;
    }
}

// ---------------------------------------------------------------------------
extern "C" void kernel_launch(void* const* d_in, const int* in_sizes, int n_in,
                              void* d_out, int out_size, void* d_ws, size_t ws_size,
                              hipStream_t stream) {
    (void)in_sizes; (void)n_in; (void)out_size; (void)ws_size;
    const float* maps = (const float*)d_in[0];
    const int*   idx1 = (const int*)d_in[1];
    const float* w1nb = (const float*)d_in[2];
    const int*   idx2 = (const int*)d_in[3];
    const float* w2nb = (const float*)d_in[4];
    const float* wd1  = (const float*)d_in[5];
    const float* wp1  = (const float*)d_in[6];
    const float* b1   = (const float*)d_in[7];
    const float* g1   = (const float*)d_in[8];
    const float* be1  = (const float*)d_in[9];
    const float* wd2  = (const float*)d_in[10];
    const float* wp2  = (const float*)d_in[11];
    const float* b2   = (const float*)d_in[12];
    const float* g2   = (const float*)d_in[13];
    const float* be2  = (const float*)d_in[14];
    float* out = (float*)d_out;

    float* bufA = (float*)d_ws;                       // 25 MB  T ping
    float* bufB = bufA + (size_t)N_PIX * FCH;         // 25 MB  T pong
    float* Yb   = bufB + (size_t)N_PIX * FCH;         // 50 MB  Y[N,64]

    const dim3 gBlk(256);
    const dim3 gGrd(N_PIX / 8);        // 8 waves/block, 1 node/wave
    const dim3 pGrd(N_PIX / 128);      // 8 waves/block, 16 nodes/wave

    // ---- Layer 1: K=6, nb=8. T0 = maps (read-only) ----
    cheb_first_kernel<8><<<gGrd, gBlk, 0, stream>>>(maps, idx1, w1nb, wd1, bufA, Yb);
    cheb_step_kernel<8><<<gGrd, gBlk, 0, stream>>>(bufA, maps, bufB, idx1, w1nb, wd1, 2, Yb);
    cheb_step_kernel<8><<<gGrd, gBlk, 0, stream>>>(bufB, bufA, bufA, idx1, w1nb, wd1, 3, Yb);
    cheb_step_kernel<8><<<gGrd, gBlk, 0, stream>>>(bufA, bufB, bufB, idx1, w1nb, wd1, 4, Yb);
    cheb_step_kernel<8><<<gGrd, gBlk, 0, stream>>>(bufB, bufA, bufA, idx1, w1nb, wd1, 5, Yb);
    // x1 parked in d_out (fully overwritten again by layer-2 epilogue)
    pointwise_ln_kernel<<<pGrd, gBlk, 0, stream>>>(Yb, wp1, b1, g1, be1, nullptr, out);

    // ---- Layer 2: K=10, nb=20. T0 = x1 (= d_out) ----
    cheb_first_kernel<20><<<gGrd, gBlk, 0, stream>>>(out, idx2, w2nb, wd2, bufA, Yb);
    cheb_step_kernel<20><<<gGrd, gBlk, 0, stream>>>(bufA, out,  bufB, idx2, w2nb, wd2, 2, Yb);
    cheb_step_kernel<20><<<gGrd, gBlk, 0, stream>>>(bufB, bufA, bufA, idx2, w2nb, wd2, 3, Yb);
    cheb_step_kernel<20><<<gGrd, gBlk, 0, stream>>>(bufA, bufB, bufB, idx2, w2nb, wd2, 4, Yb);
    cheb_step_kernel<20><<<gGrd, gBlk, 0, stream>>>(bufB, bufA, bufA, idx2, w2nb, wd2, 5, Yb);
    cheb_step_kernel<20><<<gGrd, gBlk, 0, stream>>>(bufA, bufB, bufB, idx2, w2nb, wd2, 6, Yb);
    cheb_step_kernel<20><<<gGrd, gBlk, 0, stream>>>(bufB, bufA, bufA, idx2, w2nb, wd2, 7, Yb);
    cheb_step_kernel<20><<<gGrd, gBlk, 0, stream>>>(bufA, bufB, bufB, idx2, w2nb, wd2, 8, Yb);
    cheb_step_kernel<20><<<gGrd, gBlk, 0, stream>>>(bufB, bufA, bufA, idx2, w2nb, wd2, 9, Yb);
    // epilogue: pointwise + relu + LN + residual(maps) -> d_out
    pointwise_ln_kernel<<<pGrd, gBlk, 0, stream>>>(Yb, wp2, b2, g2, be2, maps, out);
}